// Loss_35828617183237
// MI455X (gfx1250) — compile-verified
//
#include <hip/hip_runtime.h>
#include <hip/hip_bf16.h>
#include <math.h>

// ---------------------------------------------------------------------------
// EucContrastive-style loss, MI455X (gfx1250, wave32).
// Memory-bound reduction: ~80 MB HBM traffic -> ~3.5 us at 23.3 TB/s.
// Wave-level reduction implemented with V_WMMA_F32_16X16X4_F32 (fp32-exact).
// ---------------------------------------------------------------------------

typedef __attribute__((ext_vector_type(2))) float v2f;
typedef __attribute__((ext_vector_type(8))) float v8f;

#define LOSS_EPS 1e-6f
#define LOSS_MARGIN 1.0f
#define BLOCK_THREADS 256
#define WAVES_PER_BLOCK (BLOCK_THREADS / 32)
#define GRID_BLOCKS 4096

// Exact fp32 wave32 sum via V_WMMA_F32_16X16X4_F32.
// A per-lane = {v, 0}: A[m][0] = v_m (lanes 0-15), A[m][2] = v_{m+16} (lanes 16-31).
// B = all ones  ->  D[m][n] = v_m + v_{m+16}  (fp32 MACs, exact).
// D layout: lane L<16 holds D[0..7][L] in its 8 VGPRs; lane L>=16 holds D[8..15][L-16].
// So per-lane sum of the 8 D components gives P (lanes 0-15) / Q (lanes 16-31),
// and one xor-16 shuffle yields the full wave total in every lane.
// Requires EXEC all-ones: call only from uniform control flow.
__device__ __forceinline__ float wave_sum_f32(float v) {
    v2f a;
    a.x = v;
    a.y = 0.0f;
    v2f b;
    b.x = 1.0f;
    b.y = 1.0f;
    v8f c = {};
    c = __builtin_amdgcn_wmma_f32_16x16x4_f32(
        /*neg_a=*/false, a, /*neg_b=*/false, b,
        /*c_mod=*/(short)0, c, /*reuse_a=*/false, /*reuse_b=*/false);
    float s = ((c[0] + c[1]) + (c[2] + c[3])) + ((c[4] + c[5]) + (c[6] + c[7]));
    s += __shfl_xor(s, 16, 32);
    return s;
}

__global__ void zero_acc_kernel(double* __restrict__ acc) {
    if (threadIdx.x < 4) acc[threadIdx.x] = 0.0;
}

__global__ __launch_bounds__(BLOCK_THREADS) void loss_main_kernel(
    const float2* __restrict__ yhat,   // [N] rows of 2 floats
    const int*    __restrict__ ytrue,  // [N]
    const int*    __restrict__ is1,    // [S] same-pair lhs
    const int*    __restrict__ is2,    // [S] same-pair rhs
    const int*    __restrict__ id1,    // [S] diff-pair lhs
    const int*    __restrict__ id2,    // [S] diff-pair rhs
    int N, int S,
    double* __restrict__ acc)          // acc[0]=bce_sum, acc[1]=mse_sum, acc[2]=con_sum
{
    const int tid    = blockIdx.x * blockDim.x + threadIdx.x;
    const int stride = gridDim.x * blockDim.x;

    float bce = 0.0f, mse = 0.0f, con = 0.0f;

    // --- Phase 1: BCE + MSE, streaming coalesced over N rows -------------
    for (int i = tid; i < N; i += stride) {
        // speculative prefetch of the next grid-stride chunk (global_prefetch_b8)
        __builtin_prefetch(yhat + i + stride, 0, 0);
        float2 p = yhat[i];
        int    c = ytrue[i];
        float  pc = (c == 0) ? p.x : p.y;  // probability of the true class
        float  po = (c == 0) ? p.y : p.x;  // probability of the other class
        // one-hot BCE row: log(p_c) + log1p(-p_other)  (negated in finalize)
        bce += logf(pc) + log1pf(-po);
        float t0 = (c == 0) ? 1.0f : 0.0f;
        float d0 = p.x - t0;
        float d1 = p.y - (1.0f - t0);
        mse += d0 * d0 + d1 * d1;
    }

    // --- Phase 2: same-class pairs, label=0 -> contribution = d^2 --------
    // d^2 = sum((diff+eps)^2); no sqrt needed.
    for (int j = tid; j < S; j += stride) {
        float2 a = yhat[is1[j]];
        float2 b = yhat[is2[j]];
        float e0 = a.x - b.x + LOSS_EPS;
        float e1 = a.y - b.y + LOSS_EPS;
        con += e0 * e0 + e1 * e1;
    }

    // --- Phase 3: diff-class pairs, label=1 -> max(margin - d, 0)^2 ------
    for (int j = tid; j < S; j += stride) {
        float2 a = yhat[id1[j]];
        float2 b = yhat[id2[j]];
        float e0 = a.x - b.x + LOSS_EPS;
        float e1 = a.y - b.y + LOSS_EPS;
        float d  = sqrtf(e0 * e0 + e1 * e1);
        float m  = fmaxf(LOSS_MARGIN - d, 0.0f);
        con += m * m;
    }

    // --- Wave32 reduction via fp32 WMMA (uniform control flow here) ------
    bce = wave_sum_f32(bce);
    mse = wave_sum_f32(mse);
    con = wave_sum_f32(con);

    // --- Cross-wave reduction in LDS, then fp64 global atomics -----------
    __shared__ float sb[3][WAVES_PER_BLOCK];
    const int wave = threadIdx.x >> 5;
    const int lane = threadIdx.x & 31;
    if (lane == 0) {
        sb[0][wave] = bce;
        sb[1][wave] = mse;
        sb[2][wave] = con;
    }
    __syncthreads();
    if (threadIdx.x == 0) {
        double b = 0.0, m = 0.0, cn = 0.0;
        #pragma unroll
        for (int w = 0; w < WAVES_PER_BLOCK; ++w) {
            b  += (double)sb[0][w];
            m  += (double)sb[1][w];
            cn += (double)sb[2][w];
        }
        atomicAdd(&acc[0], b);   // global_atomic_add_f64
        atomicAdd(&acc[1], m);
        atomicAdd(&acc[2], cn);
    }
}

__global__ void finalize_kernel(const double* __restrict__ acc,
                                float* __restrict__ out, int N, int S) {
    double nc  = (double)N * 2.0;             // N * C entries for bce/mse means
    double bce = -acc[0] / nc;
    double mse =  acc[1] / nc;
    double con =  acc[2] / (2.0 * (double)S); // mean over 2S pairs
    out[0] = (float)(bce + mse + con);
}

extern "C" void kernel_launch(void* const* d_in, const int* in_sizes, int n_in,
                              void* d_out, int out_size, void* d_ws, size_t ws_size,
                              hipStream_t stream) {
    const float2* yhat  = (const float2*)d_in[0];
    const int*    ytrue = (const int*)d_in[1];
    const int*    is1   = (const int*)d_in[2];
    const int*    is2   = (const int*)d_in[3];
    const int*    id1   = (const int*)d_in[4];
    const int*    id2   = (const int*)d_in[5];

    const int N = in_sizes[1];  // rows (y_true length)
    const int S = in_sizes[2];  // same-pair count (== diff-pair count)

    double* acc = (double*)d_ws;

    zero_acc_kernel<<<1, 32, 0, stream>>>(acc);
    loss_main_kernel<<<GRID_BLOCKS, BLOCK_THREADS, 0, stream>>>(
        yhat, ytrue, is1, is2, id1, id2, N, S, acc);
    finalize_kernel<<<1, 1, 0, stream>>>(acc, (float*)d_out, N, S);
}